// CppGraphModule_67388036874281
// MI455X (gfx1250) — compile-verified
//
#include <hip/hip_runtime.h>
#include <math.h>

#define EPSF 1e-10f

// Matches: clip(+-1e6) then where(isfinite, v, 0). NaN must map to 0, so test first.
__device__ __forceinline__ float guardf(float v) {
    if (!isfinite(v)) return 0.0f;
    return fminf(fmaxf(v, -1.0e6f), 1.0e6f);
}
__device__ __forceinline__ float sgnf(float v) {
    return (v > 0.0f) ? 1.0f : ((v < 0.0f) ? -1.0f : 0.0f);
}
__device__ __forceinline__ float safe_divf(float l, float r) {
    return l / (fabsf(r) + EPSF) * sgnf(r);
}
// x^p for x > 0 via hardware log2/exp2 (v_log_f32 / v_exp_f32).
__device__ __forceinline__ float pow_pos(float x, float p) {
    return __builtin_amdgcn_exp2f(p * __builtin_amdgcn_logf(x));
}

// CDNA5 async global->LDS (ASYNCcnt-tracked), GVS mode: saddr(64b SGPR pair) + vaddr(32b byte offset)
#define ASYNC_LOAD_B128(lds_off, gbyte_off, base_ptr)                         \
    asm volatile("global_load_async_to_lds_b128 %0, %1, %2"                   \
                 :: "v"(lds_off), "v"(gbyte_off), "s"(base_ptr) : "memory")
#define WAIT_ASYNC_LE_1() asm volatile("s_wait_asynccnt 0x1" ::: "memory")
#define WAIT_ASYNC_LE_0() asm volatile("s_wait_asynccnt 0x0" ::: "memory")

__global__ __launch_bounds__(256) void graph_eval_kernel(
    const float* __restrict__ x, const float* __restrict__ w,
    const float* __restrict__ bias, float* __restrict__ out,
    int n, int ntiles)
{
    __shared__ float4 buf[2][256];
    const int tid = threadIdx.x;
    // addrspace(3) offset = low 32 bits of the generic (flat) pointer
    const unsigned lds0 = (unsigned)(size_t)&buf[0][tid];
    const unsigned lds1 = (unsigned)(size_t)&buf[1][tid];

    // Uniform weights -> scalar loads through the constant path
    float wv[19];
#pragma unroll
    for (int i = 0; i < 19; ++i) wv[i] = w[i];
    const float wb = bias[0];

    int tile = blockIdx.x;
    if (tile >= ntiles) return;

    // Prologue: stage this block's first tile (each lane lands its row's 16B: features 0..3)
    {
        int row = tile * 256 + tid; if (row >= n) row = n - 1;
        ASYNC_LOAD_B128(lds0, (unsigned)row * 32u, x);
    }

    int parity = 0;
    for (; tile < ntiles; tile += (int)gridDim.x) {
        const int next = tile + (int)gridDim.x;
        if (next < ntiles) {
            int row = next * 256 + tid; if (row >= n) row = n - 1;
            ASYNC_LOAD_B128(parity ? lds0 : lds1, (unsigned)row * 32u, x);
            WAIT_ASYNC_LE_1();   // current tile landed; next still in flight
        } else {
            WAIT_ASYNC_LE_0();
        }
        // Each lane reads back only its own staged row -> no barrier needed.
        const float4 f = buf[parity][tid];

        // ---- static DAG, fully unrolled (nodes 0..18) ----
        const float s0 = guardf(f.x);
        const float s1 = guardf(f.y);
        const float s2 = guardf(f.z);
        const float s3 = guardf(f.w);
        const float s4 = 2.5f;
        // n5: 1.1*sin(1.3*x0 + 0.2)
        const float s5 = guardf(1.1f * sinf(fmaf(1.3f, s0, 0.2f)));
        // n6: signed-pow p=2.0 (even) -> (|x1|+eps)^2
        const float t6 = fabsf(s1) + EPSF;
        const float s6 = guardf(t6 * t6);
        // n7: signed-pow p=1.7 -> sign(x2)*(|x2|+eps)^1.7 ; base > 0 always
        const float t7 = fabsf(s2) + EPSF;
        const float s7 = guardf(sgnf(s2) * pow_pos(t7, 1.7f));
        // n8: exp(clip(0.5*x3, -20, 20))
        const float t8 = fminf(fmaxf(0.5f * s3, -20.0f), 20.0f);
        const float s8 = guardf(expf(t8));
        // n9: log(|x0|+eps)
        const float s9 = guardf(logf(fabsf(s0) + EPSF));
        // n10: x1^3
        const float s10 = guardf(s1 * s1 * s1);
        // n11..n15
        const float s11 = guardf(s5 + s6);
        const float s12 = guardf(s7 - s8);
        const float s13 = guardf(s9 * s10);
        const float s14 = guardf(safe_divf(s11, s4));
        const float s15 = guardf(safe_divf(s12, s13));
        // n16: softmax-combine, tau = 0.5
        const float m  = fmaxf(s14, s15);
        const float el = expf((s14 - m) * 2.0f);
        const float er = expf((s15 - m) * 2.0f);
        const float s16 = guardf((s14 * el + s15 * er) / (el + er));
        // n17: 0.9*sin(0.7*n16 - 0.3)
        const float s17 = guardf(0.9f * sinf(fmaf(0.7f, s16, -0.3f)));
        // n18: n17 + n11
        const float s18 = guardf(s17 + s11);

        float acc = wb;
        acc = fmaf(wv[0],  s0,  acc);
        acc = fmaf(wv[1],  s1,  acc);
        acc = fmaf(wv[2],  s2,  acc);
        acc = fmaf(wv[3],  s3,  acc);
        acc = fmaf(wv[4],  s4,  acc);
        acc = fmaf(wv[5],  s5,  acc);
        acc = fmaf(wv[6],  s6,  acc);
        acc = fmaf(wv[7],  s7,  acc);
        acc = fmaf(wv[8],  s8,  acc);
        acc = fmaf(wv[9],  s9,  acc);
        acc = fmaf(wv[10], s10, acc);
        acc = fmaf(wv[11], s11, acc);
        acc = fmaf(wv[12], s12, acc);
        acc = fmaf(wv[13], s13, acc);
        acc = fmaf(wv[14], s14, acc);
        acc = fmaf(wv[15], s15, acc);
        acc = fmaf(wv[16], s16, acc);
        acc = fmaf(wv[17], s17, acc);
        acc = fmaf(wv[18], s18, acc);

        const int row = tile * 256 + tid;
        if (row < n) out[row] = acc;
        parity ^= 1;
    }
}

extern "C" void kernel_launch(void* const* d_in, const int* in_sizes, int n_in,
                              void* d_out, int out_size, void* d_ws, size_t ws_size,
                              hipStream_t stream) {
    (void)n_in; (void)d_ws; (void)ws_size;
    const float* x    = (const float*)d_in[0];
    const float* w    = (const float*)d_in[1];
    const float* bias = (const float*)d_in[2];
    float* out = (float*)d_out;

    const int n = in_sizes[0] / 8;          // rows (2,000,000)
    (void)out_size;
    const int ntiles = (n + 255) / 256;
    int blocks = ntiles < 2048 ? ntiles : 2048;  // persistent blocks, grid-stride pipeline

    hipLaunchKernelGGL(graph_eval_kernel, dim3(blocks), dim3(256), 0, stream,
                       x, w, bias, out, n, ntiles);
}